// InstanceLossBoost_83124797047544
// MI455X (gfx1250) — compile-verified
//
#include <hip/hip_runtime.h>
#include <stdint.h>

// ---------------------------------------------------------------------------
// InstanceLossBoost for MI455X (gfx1250).
//
// Reference: per-row argmax/max over c[B,128], per-class top-P selection by
// confidence, merge with current labels, mask conf < ALPHA to -1.
// Replaced the reference's global lexsort with per-class confidence
// histograms over [ALPHA, 1] (exact to one histogram bin of resolution;
// samples below ALPHA can never outrank a survivor, so they need no bins).
//
// Data movement: Tensor Data Mover (tensor_load_to_lds) streams 128-row
// tiles of c into LDS, double buffered, with TDM padding (4 dwords per
// 128 dwords) giving a 132-float row pitch -> conflict-free ds_load_b128.
// ---------------------------------------------------------------------------

#define KCLS       128
#define TILE_ROWS  128
#define ROW_PITCH  132          // 128 data floats + 4 TDM pad floats
#define TPB        8            // tiles per block
#define NBINS      4096
#define ALPHA_F    0.99f

typedef __attribute__((ext_vector_type(4))) unsigned int v4u;
typedef __attribute__((ext_vector_type(8))) int          v8i;
typedef __attribute__((ext_vector_type(4))) int          v4i;

// Monotone confidence -> bin. Bin 0 = below ALPHA (never histogrammed,
// never selectable). Bins 1..4095 cover [ALPHA, 1.0].
__device__ __forceinline__ int bin_of(float cf) {
    if (cf < ALPHA_F) return 0;
    int b = (int)((cf - ALPHA_F) * (4094.0f / (1.0f - ALPHA_F)));
    if (b > NBINS - 2) b = NBINS - 2;
    return 1 + b;
}

#if __has_builtin(__builtin_amdgcn_tensor_load_to_lds)
#define HAVE_TDM 1
#else
#define HAVE_TDM 0
#endif

// Issue one TDM descriptor: copy nElems consecutive f32 from gsrc into LDS
// at byte offset ldsOff, inserting 4 pad dwords after every 128 dwords
// (pad_interval code 6 = 128 dwords, pad_amount code 3 = 4 dwords).
__device__ __forceinline__ void tdm_load_tile(const float* gsrc,
                                              unsigned ldsOff, int nElems) {
#if HAVE_TDM
    unsigned long long ga = (unsigned long long)(uintptr_t)gsrc;
    v4u g0;
    g0.x = 1u;                                        // count=1, user desc
    g0.y = ldsOff;                                    // lds_addr
    g0.z = (unsigned)(ga & 0xFFFFFFFFu);              // global_addr[31:0]
    g0.w = (unsigned)((ga >> 32) & 0x01FFFFFFu)       // global_addr[56:32]
         | (2u << 30);                                // type = 2 (image)
    v8i g1;
    g1[0] = (int)((2u << 16)     // data_size = 4 bytes
                | (1u << 20)     // pad_enable
                | (6u << 22)     // pad_interval: 128 dwords
                | (3u << 25));   // pad_amount: 4 dwords
    g1[1] = (int)(((unsigned)nElems & 0xFFFFu) << 16);          // tensor_dim0 lo
    g1[2] = (int)(((unsigned)nElems >> 16) | (1u << 16));       // dim0 hi | tensor_dim1=1
    g1[3] = (int)(((unsigned)nElems & 0xFFFFu) << 16);          // tile_dim0
    g1[4] = 0;                                                  // tile_dim1/2 unused
    g1[5] = nElems;                                             // dim0 stride (unused for 1D)
    g1[6] = 0;
    g1[7] = 0;
    v4i gz4 = {0, 0, 0, 0};                // groups 2/3 disabled (<=2D tensor)
    v8i gz8 = {0, 0, 0, 0, 0, 0, 0, 0};   // extra group (clang-23 6-arg form)
    __builtin_amdgcn_tensor_load_to_lds(g0, g1, gz4, gz4, gz8, 0);
#else
    (void)gsrc; (void)ldsOff; (void)nElems;
#endif
}

__device__ __forceinline__ void wait_tensor(int n) {
#if HAVE_TDM
    if (n == 0) __builtin_amdgcn_s_wait_tensorcnt(0);
    else        __builtin_amdgcn_s_wait_tensorcnt(1);
#else
    (void)n;
#endif
}

// ---------------------------------------------------------------------------
// Pass 0: zero the class x bin histogram.
// ---------------------------------------------------------------------------
__global__ void ilb_zero(unsigned* __restrict__ hist, int n) {
    int i = blockIdx.x * blockDim.x + threadIdx.x;
    if (i < n) hist[i] = 0u;
}

// ---------------------------------------------------------------------------
// Pass 1: TDM-staged streaming argmax/max + per-(class,bin) histogram.
// 128 threads = 4 waves; each wave TDM-copies its 32-row slice; each thread
// reduces one 128-float row from LDS (32x float4 ds_load_b128, strict '>'
// matches jnp.argmax first-max tie rule).
// ---------------------------------------------------------------------------
__global__ __launch_bounds__(TILE_ROWS)
void ilb_pass1(const float* __restrict__ c,
               float* __restrict__ conf,
               unsigned char* __restrict__ pred,
               unsigned* __restrict__ hist,
               int B) {
    __shared__ float smem[2 * TILE_ROWS * ROW_PITCH];   // 135168 B

    const int t = threadIdx.x;
    const int wave = __builtin_amdgcn_readfirstlane(t >> 5);   // SGPR wave id
    const unsigned smemBase = (unsigned)(uintptr_t)(void*)smem;
    const long long blockRow0 = (long long)blockIdx.x * (TILE_ROWS * TPB);

#if !HAVE_TDM
    // Fallback path (no TDM builtin): direct cooperative loads into LDS.
    for (int tt = 0; tt < TPB; ++tt) {
        long long r0 = blockRow0 + (long long)tt * TILE_ROWS;
        for (int r = 0; r < TILE_ROWS; ++r) {
            long long row = r0 + r;
            if (row < B && t < KCLS) smem[r * ROW_PITCH + t] = c[row * KCLS + t];
        }
        __syncthreads();
        long long row = r0 + t;
        if (row < B) {
            const float* rp = &smem[t * ROW_PITCH];
            float best = -1.0f; int bi = 0;
            for (int j = 0; j < KCLS; ++j) {
                float v = rp[j];
                if (v > best) { best = v; bi = j; }
            }
            conf[row] = best;
            pred[row] = (unsigned char)bi;
            int b = bin_of(best);
            if (b > 0) atomicAdd(&hist[bi * NBINS + b], 1u);
        }
        __syncthreads();
    }
#else
    // Prologue: issue tile 0 (each wave: its 32 rows = 4096 f32 = 16 KB).
    if (blockRow0 < B) {
        const float* g = c + (blockRow0 + (long long)wave * 32) * KCLS;
        tdm_load_tile(g, smemBase + (unsigned)(wave * 32 * ROW_PITCH * 4), 32 * KCLS);
    }

    for (int tt = 0; tt < TPB; ++tt) {
        const int cur = tt & 1;
        const int nxt = cur ^ 1;
        long long rNext = blockRow0 + (long long)(tt + 1) * TILE_ROWS;
        if ((tt + 1 < TPB) && (rNext < B)) {
            const float* g = c + (rNext + (long long)wave * 32) * KCLS;
            tdm_load_tile(g,
                          smemBase + (unsigned)((nxt * TILE_ROWS + wave * 32) * ROW_PITCH * 4),
                          32 * KCLS);
            wait_tensor(1);          // tile tt landed (per-wave, in-order)
        } else {
            wait_tensor(0);
        }
        __syncthreads();             // all 4 waves' slices confirmed

        long long row = blockRow0 + (long long)tt * TILE_ROWS + t;
        if (row < B) {
            const float* rp = &smem[(cur * TILE_ROWS + t) * ROW_PITCH];
            float best = -1.0f; int bi = 0;
#pragma unroll
            for (int m = 0; m < KCLS / 4; ++m) {
                float4 v = *(const float4*)(rp + 4 * m);
                if (v.x > best) { best = v.x; bi = 4 * m + 0; }
                if (v.y > best) { best = v.y; bi = 4 * m + 1; }
                if (v.z > best) { best = v.z; bi = 4 * m + 2; }
                if (v.w > best) { best = v.w; bi = 4 * m + 3; }
            }
            conf[row] = best;
            pred[row] = (unsigned char)bi;
            int b = bin_of(best);
            if (b > 0) atomicAdd(&hist[bi * NBINS + b], 1u);
        }
        __syncthreads();             // reads done before buffer reuse
    }
#endif
}

// ---------------------------------------------------------------------------
// Pass 2: per class, suffix-scan bins from the top; A[k] = lowest bin such
// that the count of samples in bins >= A[k] stays <= P.
// ---------------------------------------------------------------------------
__global__ void ilb_pass2(const unsigned* __restrict__ hist,
                          int* __restrict__ A, int P) {
    int k = threadIdx.x;
    if (k < KCLS) {
        const unsigned* h = hist + (size_t)k * NBINS;
        unsigned s = 0;
        int a = 1;
        for (int b = NBINS - 1; b >= 1; --b) {
            unsigned ns = s + h[b];
            if (ns > (unsigned)P) { a = b + 1; break; }
            s = ns;
        }
        A[k] = a;
    }
}

// ---------------------------------------------------------------------------
// Pass 3: finalize labels + pass index through.
// d_out = [ final[B] : int32 | index[B] : int32 ]
// ---------------------------------------------------------------------------
__global__ void ilb_pass3(const float* __restrict__ conf,
                          const unsigned char* __restrict__ pred,
                          const int* __restrict__ A,
                          const int* __restrict__ cur,
                          const int* __restrict__ index,
                          int* __restrict__ out, int B) {
    int i = blockIdx.x * blockDim.x + threadIdx.x;
    if (i < B) {
        float cf = conf[i];
        int cu = cur[i];
        int o;
        if (cf < ALPHA_F) {
            o = -1;
        } else if (cu != -1) {
            o = cu;
        } else {
            int p = (int)pred[i];
            o = (bin_of(cf) >= A[p]) ? p : -1;
        }
        out[i] = o;
        out[B + i] = index[i];
    }
}

// ---------------------------------------------------------------------------
extern "C" void kernel_launch(void* const* d_in, const int* in_sizes, int n_in,
                              void* d_out, int out_size, void* d_ws, size_t ws_size,
                              hipStream_t stream) {
    const float* c          = (const float*)d_in[0];
    const int*   pl_cur     = (const int*)d_in[1];
    const int*   index      = (const int*)d_in[2];
    int*         out        = (int*)d_out;           // int32 outputs, bit-exact view
    const int B = in_sizes[1];

    // Workspace layout (bytes):
    //   [0, 2 MiB)            hist: KCLS * NBINS u32
    //   [2 MiB, +512)         A:    KCLS int
    //   [.., +4B)             conf: B float
    //   [.., +1B)             pred: B u8
    unsigned char* ws = (unsigned char*)d_ws;
    unsigned*      hist = (unsigned*)ws;
    size_t off = (size_t)KCLS * NBINS * sizeof(unsigned);
    int*           A    = (int*)(ws + off);
    off += 512;
    float*         conf = (float*)(ws + off);
    off += (size_t)B * sizeof(float);
    unsigned char* pred = ws + off;
    (void)ws_size; (void)n_in; (void)out_size;

    const int P = (B + 2 * KCLS - 1) / (2 * KCLS);   // ceil(B/K * GAMMA), GAMMA=0.5

    // Pass 0: zero histogram.
    {
        int n = KCLS * NBINS;
        ilb_zero<<<(n + 255) / 256, 256, 0, stream>>>(hist, n);
    }
    // Pass 1: the 1 GiB streaming pass (TDM double-buffered tiles).
    {
        long long rowsPerBlock = (long long)TILE_ROWS * TPB;
        int nBlocks = (int)((B + rowsPerBlock - 1) / rowsPerBlock);
        ilb_pass1<<<nBlocks, TILE_ROWS, 0, stream>>>(c, conf, pred, hist, B);
    }
    // Pass 2: per-class cut bin.
    ilb_pass2<<<1, KCLS, 0, stream>>>(hist, A, P);
    // Pass 3: finalize.
    ilb_pass3<<<(B + 255) / 256, 256, 0, stream>>>(conf, pred, A, pl_cur, index, out, B);
}